// InteractionBlock_Physnet_17257178595652
// MI455X (gfx1250) — compile-verified
//
#include <hip/hip_runtime.h>

// PhysNet interaction block for MI455X (gfx1250, wave32, WMMA + TDM).
// Outputs: x [N,256] then g [E,256], concatenated in d_out.
//
// Roofline: ~330 GFLOP fp32 over ~1.5-2 GB of HBM traffic; y (102MB) and the
// scatter target m (102MB) are L2-resident (192MB L2), so the edge gather and
// the 205M fp32 atomics stay on-chip. Dense GEMMs use V_WMMA_F32_16X16X4_F32.
// Weight tiles are staged into LDS by the Tensor Data Mover, double-buffered
// so the DMA of chunk c+1 overlaps WMMA on chunk c (TENSORcnt <= 1 wait).

typedef __attribute__((ext_vector_type(2))) float v2f;
typedef __attribute__((ext_vector_type(8))) float v8f;
typedef unsigned int u32;
typedef unsigned long long u64;
typedef __attribute__((ext_vector_type(4))) u32 v4u;
typedef __attribute__((ext_vector_type(8))) int v8i;
typedef __attribute__((ext_vector_type(4))) int v4i;

#define FDIM 256
#define NT 8               // 8 n-tiles of 16 -> 128 cols per wave
#define WPB 4              // waves per block (128 threads)
#define KC 64              // K-chunk staged in LDS per TDM transfer
#define LDSB_STRIDE 132    // 128 cols + 4 floats TDM pad (16B per 512B row)

static __device__ __forceinline__ v8f wmma4(v2f a, v2f b, v8f c) {
    // V_WMMA_F32_16X16X4_F32: D = A(16x4) * B(4x16) + C(16x16), fp32
    return __builtin_amdgcn_wmma_f32_16x16x4_f32(
        false, a, false, b, (short)0, c, false, false);
}

// TDM: DMA a rows x 128 fp32 tile of W (row stride 256 floats) into LDS at
// ldsAddr, inserting 16B padding after each 512B row (-> stride 132 floats).
// D# built per CDNA5 ISA 8.3-8.6. Issued by one wave; EXEC is ignored by TDM.
static __device__ __forceinline__ void tdm_load_tile(const float* gsrc, int rows,
                                                     u32 ldsAddr) {
    u64 ga = (u64)(uintptr_t)gsrc;
    v4u g0;
    g0.x = 1u;                                           // count=1, no gather
    g0.y = ldsAddr;                                      // lds_addr [63:32]
    g0.z = (u32)ga;                                      // global_addr[31:0]
    g0.w = ((u32)(ga >> 32) & 0x01FFFFFFu) | (2u << 30); // addr[56:32] | type=2
    v8i g1;
    g1[0] = (int)((2u << 16)      // data_size = 4B
                | (1u << 20)      // pad_enable
                | (6u << 22)      // pad_interval: 128 DWORDs (2^(6+1))
                | (3u << 25));    // pad_amount: 4 DWORDs (3+1)
    g1[1] = (int)(128u << 16);                // tensor_dim0 = 128
    g1[2] = (int)((u32)rows << 16);           // tensor_dim1[15:0] = rows
    g1[3] = (int)(128u << 16);                // tile_dim0 = 128
    g1[4] = (int)((u32)rows & 0xFFFFu);       // tile_dim1 = rows, tile_dim2 = 0
    g1[5] = 256;                              // tensor_dim0_stride = 256 elements
    g1[6] = 0;
    g1[7] = 0;
    v4i g2 = {0, 0, 0, 0};                    // 2-D tile: higher dims unused
    v4i g3 = {0, 0, 0, 0};
#if __clang_major__ >= 23
    v8i g4 = {0, 0, 0, 0, 0, 0, 0, 0};
    __builtin_amdgcn_tensor_load_to_lds(g0, g1, g2, g3, g4, 0);
#else
    __builtin_amdgcn_tensor_load_to_lds(g0, g1, g2, g3, 0);
#endif
}

// One K-step (4 deep): 8 WMMAs sharing one A fragment, B from LDS.
static __device__ __forceinline__ void kstep(const float* aK, const float* bbuf,
                                             int k0, int half, int l16, v8f acc[NT]) {
    v2f a = *(const v2f*)(aK + k0);
    const float* bk = bbuf + (size_t)(k0 + 2 * half) * LDSB_STRIDE + l16;
#pragma unroll
    for (int j = 0; j < NT; ++j) {
        v2f b;
        b.x = bk[j * 16];
        b.y = bk[j * 16 + LDSB_STRIDE];
        acc[j] = wmma4(a, b, acc[j]);
    }
}

// Shared mainloop: acc[NT] += A[16 rows x K] @ W[K x 128 cols] using WMMA with
// double-buffered TDM-staged B tiles. All waves must enter (barriers inside).
static __device__ __forceinline__ void gemm_mainloop(
    const float* __restrict__ A, const float* __restrict__ W,
    float (*ldsB)[KC * LDSB_STRIDE], int M, int K, int rowBase, int nBase,
    int lane, int wave, v8f acc[NT])
{
    const int half = lane >> 4;   // lanes 0-15 vs 16-31 (K-pair select)
    const int l16  = lane & 15;

    int aRow = rowBase + l16;
    if (aRow >= M) aRow = M - 1;                 // clamp loads; stores masked later
    const float* aPtr = A + (size_t)aRow * K + 2 * half;
    const u32 lds0 = (u32)(uintptr_t)&ldsB[0][0];   // low 32 bits = LDS offset
    const u32 lds1 = (u32)(uintptr_t)&ldsB[1][0];

    if (wave == 0) {                              // prologue: start chunk 0 DMA
        const int rows0 = K < KC ? K : KC;
        tdm_load_tile(W + nBase, rows0, lds0);
    }

    int buf = 0;
    for (int kc = 0; kc < K; kc += KC, buf ^= 1) {
        const int rows = (K - kc) < KC ? (K - kc) : KC;
        if (wave == 0) {
            const int kn = kc + KC;
            if (kn < K) {                         // prefetch next chunk, overlap
                const int rowsN = (K - kn) < KC ? (K - kn) : KC;
                tdm_load_tile(W + (size_t)kn * FDIM + nBase, rowsN,
                              buf ? lds0 : lds1);
                __builtin_amdgcn_s_wait_tensorcnt(1);  // current chunk resident
            } else {
                __builtin_amdgcn_s_wait_tensorcnt(0);
            }
        }
        __syncthreads();
        const float* bbuf = &ldsB[buf][0];
        const float* aK = aPtr + kc;
        if (rows == KC) {                         // constant trip: dense WMMA body
#pragma unroll 4
            for (int k0 = 0; k0 < KC; k0 += 4) kstep(aK, bbuf, k0, half, l16, acc);
        } else {                                  // K tail (e.g. 36 rows of K=100)
            for (int k0 = 0; k0 < rows; k0 += 4) kstep(aK, bbuf, k0, half, l16, acc);
        }
        __syncthreads();                          // WAR before buffer reuse
    }
}

// Generic GEMM: out[M,256] = A[M,K] @ W[K,256] + bias (+ epilogue)
// mode 0: out = acc + bias
// mode 1: out = acc + bias + add[row,col]            (residual; add may alias out)
// mode 2: out = acc + bias + u[col] * add[row,col]   (u*h skip connection)
__global__ __launch_bounds__(128) void gemm_f32_wmma(
    const float* __restrict__ A, const float* __restrict__ W,
    const float* __restrict__ bias, const float* add, const float* __restrict__ u,
    float* out, int M, int K, int mode)
{
    __shared__ float ldsB[2][KC * LDSB_STRIDE];
    if (blockDim.y == 777u) {                     // opaque: keep LDS "written"
        ldsB[0][threadIdx.x] = 0.f;
        ldsB[1][threadIdx.x] = 0.f;
    }

    const int lane = threadIdx.x & 31;
    const int wave = threadIdx.x >> 5;
    const int rowBase = (blockIdx.x * WPB + wave) * 16;
    const int nBase = blockIdx.y * (NT * 16);
    const int half = lane >> 4;
    const int l16  = lane & 15;

    v8f acc[NT];
    const v8f vzero = {0.f,0.f,0.f,0.f,0.f,0.f,0.f,0.f};
#pragma unroll
    for (int j = 0; j < NT; ++j) acc[j] = vzero;

    gemm_mainloop(A, W, ldsB, M, K, rowBase, nBase, lane, wave, acc);

    // C layout: VGPR i -> row i (lanes 0-15) / i+8 (lanes 16-31); col = lane%16.
#pragma unroll
    for (int i = 0; i < 8; ++i) {
        int row = rowBase + i + 8 * half;
        if (row >= M) continue;
        size_t base = (size_t)row * FDIM;
#pragma unroll
        for (int j = 0; j < NT; ++j) {
            int col = nBase + j * 16 + l16;
            float v = acc[j][i] + bias[col];
            if (mode == 1)      v += add[base + col];
            else if (mode == 2) v += u[col] * add[base + col];
            out[base + col] = v;
        }
    }
}

// Fused edge kernel:
//   g[e,:]   = edge_attr[e,:] @ W_d2f + b_d2f      (WMMA, K=DF=100)
//   m[dst,:] += g[e,:] * y[src,:]                  (L2 gather + fp32 atomics)
__global__ __launch_bounds__(128) void edge_gemm_scatter(
    const float* __restrict__ EA, const float* __restrict__ W,
    const float* __restrict__ bias,
    const int* __restrict__ srcIdx, const int* __restrict__ dstIdx,
    const float* __restrict__ y,
    float* __restrict__ outG, float* m,
    int E, int K)
{
    __shared__ float ldsB[2][KC * LDSB_STRIDE];
    if (blockDim.y == 777u) {
        ldsB[0][threadIdx.x] = 0.f;
        ldsB[1][threadIdx.x] = 0.f;
    }

    const int lane = threadIdx.x & 31;
    const int wave = threadIdx.x >> 5;
    const int rowBase = (blockIdx.x * WPB + wave) * 16;
    const int nBase = blockIdx.y * (NT * 16);
    const int half = lane >> 4;
    const int l16  = lane & 15;

    v8f acc[NT];
    const v8f vzero = {0.f,0.f,0.f,0.f,0.f,0.f,0.f,0.f};
#pragma unroll
    for (int j = 0; j < NT; ++j) acc[j] = vzero;

    gemm_mainloop(EA, W, ldsB, E, K, rowBase, nBase, lane, wave, acc);

#pragma unroll
    for (int i = 0; i < 8; ++i) {
        int edge = rowBase + i + 8 * half;
        if (edge >= E) continue;
        int s = srcIdx[edge];
        int d = dstIdx[edge];
        size_t gBase = (size_t)edge * FDIM;
        size_t sBase = (size_t)s * FDIM;
        size_t dBase = (size_t)d * FDIM;
#pragma unroll
        for (int j = 0; j < NT; ++j) {
            int col = nBase + j * 16 + l16;
            float g = acc[j][i] + bias[col];
            outG[gBase + col] = g;                  // g output
            float msg = g * y[sBase + col];         // gather (L2-resident)
            unsafeAtomicAdd(&m[dBase + col], msg);  // global_atomic_add_f32
        }
    }
}

extern "C" void kernel_launch(void* const* d_in, const int* in_sizes, int n_in,
                              void* d_out, int out_size, void* d_ws, size_t ws_size,
                              hipStream_t stream) {
    const float* h          = (const float*)d_in[0];
    const int*   edge_index = (const int*)  d_in[1];
    // d_in[2] edge_weight: unused by reference
    const float* edge_attr  = (const float*)d_in[3];
    // d_in[4] batch: unused
    const float* W_d2f = (const float*)d_in[5];
    const float* b_d2f = (const float*)d_in[6];
    const float* W_i   = (const float*)d_in[7];
    const float* b_i   = (const float*)d_in[8];
    const float* res_W1 = (const float*)d_in[9];
    const float* res_b1 = (const float*)d_in[10];
    const float* res_W2 = (const float*)d_in[11];
    const float* res_b2 = (const float*)d_in[12];
    const float* W_out = (const float*)d_in[13];
    const float* b_out = (const float*)d_in[14];
    const float* u     = (const float*)d_in[15];
    const float* at_W1 = (const float*)d_in[16];
    const float* at_b1 = (const float*)d_in[17];
    const float* at_W2 = (const float*)d_in[18];
    const float* at_b2 = (const float*)d_in[19];

    const int N  = in_sizes[0] / FDIM;
    const int E  = in_sizes[2];
    const int DF = in_sizes[3] / E;
    const int NRES    = in_sizes[9]  / (FDIM * FDIM);
    const int NRES_AT = in_sizes[16] / (FDIM * FDIM);

    float* x_out = (float*)d_out;                 // [N,256]
    float* g_out = x_out + (size_t)N * FDIM;      // [E,256]

    float* yt = (float*)d_ws;                     // y, later reused as t [N,256]
    float* m  = yt + (size_t)N * FDIM;            // scatter accumulator [N,256]
    (void)ws_size; (void)n_in; (void)out_size;    // needs ws >= 2*N*256*4 bytes

    const dim3 blk(128);
    const int colBlocks = FDIM / (NT * 16);       // 2
    auto grid = [&](int M) { return dim3((unsigned)((M + 16 * WPB - 1) / (16 * WPB)),
                                         (unsigned)colBlocks); };

    // 1) y = h @ W_i + b_i
    gemm_f32_wmma<<<grid(N), blk, 0, stream>>>(h, W_i, b_i, nullptr, nullptr, yt,
                                               N, FDIM, 0);
    // 2) m = 0
    hipMemsetAsync(m, 0, (size_t)N * FDIM * sizeof(float), stream);
    // 3) g = edge_attr @ W_d2f + b_d2f ; m[dst] += g * y[src]
    edge_gemm_scatter<<<grid(E), blk, 0, stream>>>(edge_attr, W_d2f, b_d2f,
                                                   edge_index, edge_index + E,
                                                   yt, g_out, m, E, DF);
    // 4) interaction residual stack on m
    for (int k = 0; k < NRES; ++k) {
        gemm_f32_wmma<<<grid(N), blk, 0, stream>>>(
            m, res_W1 + (size_t)k * FDIM * FDIM, res_b1 + (size_t)k * FDIM,
            nullptr, nullptr, yt, N, FDIM, 0);
        gemm_f32_wmma<<<grid(N), blk, 0, stream>>>(
            yt, res_W2 + (size_t)k * FDIM * FDIM, res_b2 + (size_t)k * FDIM,
            m, nullptr, m, N, FDIM, 1);
    }
    // 5) x = u*h + m @ W_out + b_out
    gemm_f32_wmma<<<grid(N), blk, 0, stream>>>(m, W_out, b_out, h, u, x_out,
                                               N, FDIM, 2);
    // 6) atomic residual stack on x
    for (int k = 0; k < NRES_AT; ++k) {
        gemm_f32_wmma<<<grid(N), blk, 0, stream>>>(
            x_out, at_W1 + (size_t)k * FDIM * FDIM, at_b1 + (size_t)k * FDIM,
            nullptr, nullptr, yt, N, FDIM, 0);
        gemm_f32_wmma<<<grid(N), blk, 0, stream>>>(
            yt, at_W2 + (size_t)k * FDIM * FDIM, at_b2 + (size_t)k * FDIM,
            x_out, nullptr, x_out, N, FDIM, 1);
    }
}